// LightGCN_60851096649961
// MI455X (gfx1250) — compile-verified
//
#include <hip/hip_runtime.h>

#define EMBED_DIM 64
#define TILE 256
#define TPB_TILES 4      // tiles (of 256 edges) per block, pipelined via ASYNCcnt
#define NLAYERS 3

// Async global->LDS copy path (CDNA5 ASYNCcnt). Proven available on this
// toolchain (round-1 diagnostic originated inside this block).
#if defined(__HIP_DEVICE_COMPILE__) && __has_builtin(__builtin_amdgcn_global_load_async_to_lds_b128)
#define HAVE_ASYNC_COPY 1
#else
#define HAVE_ASYNC_COPY 0
#endif

#if HAVE_ASYNC_COPY
typedef int v4i __attribute__((ext_vector_type(4)));
typedef __attribute__((address_space(1))) v4i* g4p;  // global <4 x i32>*
typedef __attribute__((address_space(3))) v4i* l4p;  // LDS    <4 x i32>*

__device__ __forceinline__ void wait_asynccnt(int n) {
  // Builtin demands a constant integer; switch over literals so every call is
  // an ICE. Call sites are unrolled, so the switch folds to a single wait.
#if __has_builtin(__builtin_amdgcn_s_wait_asynccnt)
  switch (n) {
    case 0:  __builtin_amdgcn_s_wait_asynccnt(0); break;
    case 1:  __builtin_amdgcn_s_wait_asynccnt(1); break;
    case 2:  __builtin_amdgcn_s_wait_asynccnt(2); break;
    default: __builtin_amdgcn_s_wait_asynccnt(3); break;
  }
#else
  switch (n) {
    case 0:  asm volatile("s_wait_asynccnt 0x0" ::: "memory"); break;
    case 1:  asm volatile("s_wait_asynccnt 0x1" ::: "memory"); break;
    case 2:  asm volatile("s_wait_asynccnt 0x2" ::: "memory"); break;
    default: asm volatile("s_wait_asynccnt 0x3" ::: "memory"); break;
  }
#endif
}
#endif

__device__ __forceinline__ void atom_add_f32(float* p, float v) {
  // Hardware GLOBAL_ATOMIC_ADD_F32 (no-return form; tracked with STOREcnt)
  unsafeAtomicAdd(p, v);
}

// 4 x 256-edge COO tiles per 256-thread block. All 4 tile copies (val,row,col)
// are issued up-front as async b128 global->LDS transfers, then drained with
// s_wait_asynccnt 3/2/1/0 so later tiles stream in while earlier tiles are
// processed. Processing: 16 lanes per edge, one float4 gather of the 256B
// source row, 4 hardware f32 atomics into the destination row. The full-tile
// hot path is unrolled x4 to keep several gathers in flight per wave.
__global__ __launch_bounds__(256)
void lgcn_spmm(const float* __restrict__ vals, const int* __restrict__ rows,
               const int* __restrict__ cols, const float* __restrict__ xc,
               float* __restrict__ xn, int nnz) {
  __shared__ float s_val[TPB_TILES][TILE];
  __shared__ int   s_row[TPB_TILES][TILE];
  __shared__ int   s_col[TPB_TILES][TILE];
  const int t = threadIdx.x;
  const long blockBase = (long)blockIdx.x * (TILE * TPB_TILES);

  // ---- stage: issue all tile copies (async when the tile is fully in range)
  for (int i = 0; i < TPB_TILES; ++i) {
    const long bt = blockBase + (long)i * TILE;
    if (bt >= nnz) break;
#if HAVE_ASYNC_COPY
    if (bt + TILE <= nnz) {
      if (t < 192) {                      // waves 0..5 issue, 1 instr per tile
        const int which = t >> 6;         // 0: vals, 1: rows, 2: cols
        const int e     = (t & 63) << 2;  // 4 elements per b128
        const void* g;
        void*       l;
        if (which == 0)      { g = (const void*)(vals + bt + e); l = (void*)&s_val[i][e]; }
        else if (which == 1) { g = (const void*)(rows + bt + e); l = (void*)&s_row[i][e]; }
        else                 { g = (const void*)(cols + bt + e); l = (void*)&s_col[i][e]; }
        __builtin_amdgcn_global_load_async_to_lds_b128((g4p)g, (l4p)l, 0, 0);
      }
      continue;
    }
#endif
    for (int k = t; k < TILE; k += 256) { // tail tile: guarded plain staging
      long e = bt + k;
      s_val[i][k] = (e < nnz) ? vals[e] : 0.f;
      s_row[i][k] = (e < nnz) ? rows[e] : 0;
      s_col[i][k] = (e < nnz) ? cols[e] : 0;
    }
  }

  // ---- drain + process
  const int lane4 = (t & 15) << 2;  // 16 lanes cover one 64-float row (float4)
  const int grp   = t >> 4;         // 16 edge groups per block
#pragma unroll
  for (int i = 0; i < TPB_TILES; ++i) {
    const long bt = blockBase + (long)i * TILE;
    if (bt >= nnz) break;
#if HAVE_ASYNC_COPY
    wait_asynccnt(TPB_TILES - 1 - i);   // tile i complete (in-order returns)
#endif
    __syncthreads();                    // LDS (async or ds-store) visible to all

    auto process = [&](int k) {
      const float v = s_val[i][k];
      const int   r = s_row[i][k];
      const int   c = s_col[i][k];
      const float4 g = *(const float4*)(xc + (size_t)c * EMBED_DIM + lane4);
      float* dst = xn + (size_t)r * EMBED_DIM + lane4;
      atom_add_f32(dst + 0, v * g.x);
      atom_add_f32(dst + 1, v * g.y);
      atom_add_f32(dst + 2, v * g.z);
      atom_add_f32(dst + 3, v * g.w);
    };

    if (bt + TILE <= nnz) {
      // hot path: compile-time trip count (16), unrolled x4 for MLP
#pragma unroll 4
      for (int k = grp; k < TILE; k += 16) process(k);
    } else {
      const int kmax = (int)(nnz - bt);
      for (int k = grp; k < kmax; k += 16) process(k);
    }
  }
}

__global__ __launch_bounds__(256)
void lgcn_init(const float4* __restrict__ ue, const float4* __restrict__ ie,
               float4* __restrict__ x0, float4* __restrict__ acc,
               int nUser4, int nTot4) {
  int i = blockIdx.x * 256 + threadIdx.x;
  if (i >= nTot4) return;
  float4 v = (i < nUser4) ? ue[i] : ie[i - nUser4];
  x0[i]  = v;
  acc[i] = v;
}

__global__ __launch_bounds__(256)
void lgcn_zero(float4* __restrict__ p, int n4) {
  int i = blockIdx.x * 256 + threadIdx.x;
  if (i < n4) p[i] = make_float4(0.f, 0.f, 0.f, 0.f);
}

__global__ __launch_bounds__(256)
void lgcn_acc(float4* __restrict__ acc, const float4* __restrict__ x,
              int n4, float scale) {
  int i = blockIdx.x * 256 + threadIdx.x;
  if (i >= n4) return;
  float4 a = acc[i];
  const float4 b = x[i];
  a.x = (a.x + b.x) * scale;
  a.y = (a.y + b.y) * scale;
  a.z = (a.z + b.z) * scale;
  a.w = (a.w + b.w) * scale;
  acc[i] = a;
}

extern "C" void kernel_launch(void* const* d_in, const int* in_sizes, int n_in,
                              void* d_out, int out_size, void* d_ws, size_t ws_size,
                              hipStream_t stream) {
  const float* ue   = (const float*)d_in[0];
  const float* ie   = (const float*)d_in[1];
  const float* vals = (const float*)d_in[2];
  const int*   rows = (const int*)d_in[3];
  const int*   cols = (const int*)d_in[4];
  float* out = (float*)d_out;

  const int nUsers = in_sizes[0] / EMBED_DIM;  // 100000
  const int nItems = in_sizes[1] / EMBED_DIM;  // 150000
  const int N      = nUsers + nItems;          // 250000
  const int nnz    = in_sizes[2];              // 4000000

  float* x0 = (float*)d_ws;
  float* x1 = x0 + (size_t)N * EMBED_DIM;      // needs 2 * 64MB of ws

  const int n4     = (N * EMBED_DIM) / 4;      // float4 elements (4,000,000)
  const int nUser4 = (nUsers * EMBED_DIM) / 4;
  const int gridE  = (n4 + 255) / 256;
  const int nBlocks = (nnz + TILE * TPB_TILES - 1) / (TILE * TPB_TILES);

  lgcn_init<<<gridE, 256, 0, stream>>>((const float4*)ue, (const float4*)ie,
                                       (float4*)x0, (float4*)out, nUser4, n4);

  float* cur = x0;
  float* nxt = x1;
  for (int layer = 0; layer < NLAYERS; ++layer) {
    lgcn_zero<<<gridE, 256, 0, stream>>>((float4*)nxt, n4);
    lgcn_spmm<<<nBlocks, 256, 0, stream>>>(vals, rows, cols, cur, nxt, nnz);
    lgcn_acc<<<gridE, 256, 0, stream>>>((float4*)out, (const float4*)nxt, n4,
                                        (layer == NLAYERS - 1) ? 0.25f : 1.0f);
    float* tmp = cur; cur = nxt; nxt = tmp;
  }
}